// MGIoU2DPlus_59760174957245
// MI455X (gfx1250) — compile-verified
//
#include <hip/hip_runtime.h>
#include <hip/hip_bf16.h>

#define NBLK 2048
#define NTHR 256
#define WAVES (NTHR / 32)

// ---------------------------------------------------------------------------
// CDNA5 async-tensor staging: global_load_async_to_lds_b128 (GVS saddr form)
// + s_wait_asynccnt. INST_OFFSET advances BOTH the LDS dest and the global
// source (CDNA5 ISA), so each box stages with 4 asm ops and a
// single v_lshlrev for the 32-bit global offset.
// ---------------------------------------------------------------------------
#if defined(__gfx1250__)
#define ASYNC_STAGE 1
#endif

#if defined(ASYNC_STAGE)
#define WAIT_ASYNC_4() asm volatile("s_wait_asynccnt 4" ::: "memory")
#define WAIT_ASYNC_0() asm volatile("s_wait_asynccnt 0" ::: "memory")
#else
#define WAIT_ASYNC_4()
#define WAIT_ASYNC_0()
#endif

// slot: generic pointer to the lane's LDS slot (fallback path)
// la:   low 32 bits of that pointer == workgroup-relative LDS byte offset
// LDS layout per slot: [0..31] pred box (8 f32), [32..63] target box (8 f32)
__device__ __forceinline__ void stage_box(float* slot, unsigned la,
                                          const float* __restrict__ pred,
                                          const float* __restrict__ tgt, int idx) {
#if defined(ASYNC_STAGE)
  unsigned voff = (unsigned)idx << 5;  // 32 B per box, per-lane VGPR offset
  unsigned la32 = la + 32u;
  asm volatile("global_load_async_to_lds_b128 %0, %1, %2" ::"v"(la), "v"(voff),
               "s"(pred)
               : "memory");
  asm volatile("global_load_async_to_lds_b128 %0, %1, %2 offset:16" ::"v"(la), "v"(voff),
               "s"(pred)
               : "memory");
  asm volatile("global_load_async_to_lds_b128 %0, %1, %2" ::"v"(la32), "v"(voff),
               "s"(tgt)
               : "memory");
  asm volatile("global_load_async_to_lds_b128 %0, %1, %2 offset:16" ::"v"(la32), "v"(voff),
               "s"(tgt)
               : "memory");
#else
  const float4* gp = (const float4*)(pred + (size_t)idx * 8);
  const float4* gt = (const float4*)(tgt + (size_t)idx * 8);
  float4* s4 = (float4*)slot;
  s4[0] = gp[0];
  s4[1] = gp[1];
  s4[2] = gt[0];
  s4[3] = gt[1];
#endif
}

// fast reciprocal: single v_rcp_f32 (≈1 ulp), skips IEEE div expansion
__device__ __forceinline__ float fast_rcp(float x) {
#if defined(__AMDGCN__)
  return __builtin_amdgcn_rcpf(x);
#else
  return 1.0f / x;
#endif
}

// ---------------------------------------------------------------------------
// Per-box math
// ---------------------------------------------------------------------------

// atan2 ordering classes: (-pi,0) < 0 < (0,pi) < pi
__device__ __forceinline__ int angcls(float x, float y) {
  if (y < 0.f) return 0;
  if (y == 0.f) return (x >= 0.f) ? 1 : 3;
  return 2;
}

__device__ __forceinline__ bool ang_less(float ax, float ay, float bx, float by) {
  int ca = angcls(ax, ay);
  int cb = angcls(bx, by);
  if (ca != cb) return ca < cb;
  return (ax * by - ay * bx) > 0.f;  // both in one open half-plane: a before b iff cross>0
}

// edge normals of angle-sorted corners; only center-relative vecs needed
// (edge = cs[next]-cs[cur] = d[next]-d[cur]; centers cancel)
__device__ __forceinline__ void candidate_axes(const float x[4], const float y[4],
                                               float nx[4], float ny[4]) {
  float cx = 0.25f * (x[0] + x[1] + x[2] + x[3]);
  float cy = 0.25f * (y[0] + y[1] + y[2] + y[3]);
  float dx0 = x[0] - cx, dy0 = y[0] - cy;
  float dx1 = x[1] - cx, dy1 = y[1] - cy;
  float dx2 = x[2] - cx, dy2 = y[2] - cy;
  float dx3 = x[3] - cx, dy3 = y[3] - cy;
  // 5-comparator sorting network on 4 elems, ascending by angle (static indices
  // so everything stays in VGPRs)
#define CSWAP(ax_, ay_, bx_, by_)                         \
  {                                                       \
    if (ang_less(bx_, by_, ax_, ay_)) {                   \
      float t_;                                           \
      t_ = ax_; ax_ = bx_; bx_ = t_;                      \
      t_ = ay_; ay_ = by_; by_ = t_;                      \
    }                                                     \
  }
  CSWAP(dx0, dy0, dx1, dy1)
  CSWAP(dx2, dy2, dx3, dy3)
  CSWAP(dx0, dy0, dx2, dy2)
  CSWAP(dx1, dy1, dx3, dy3)
  CSWAP(dx1, dy1, dx2, dy2)
#undef CSWAP
  // edges -> normals (ey, -ex)
  nx[0] = dy1 - dy0; ny[0] = -(dx1 - dx0);
  nx[1] = dy2 - dy1; ny[1] = -(dx2 - dx1);
  nx[2] = dy3 - dy2; ny[2] = -(dx3 - dx2);
  nx[3] = dy0 - dy3; ny[3] = -(dx0 - dx3);
}

__device__ __forceinline__ float convexity(const float x[4], const float y[4]) {
  float cr[4];
#pragma unroll
  for (int k = 0; k < 4; ++k) {
    int p = (k + 3) & 3, q = (k + 1) & 3;
    float e1x = x[p] - x[k], e1y = y[p] - y[k];
    float e2x = x[q] - x[k], e2y = y[q] - y[k];
    cr[k] = e1x * e2y - e1y * e2x;
  }
  float c0 = cr[0];
  float sref = (c0 > 0.f) ? 1.f : ((c0 < 0.f) ? -1.f : 1.f);
  float pen = 0.f;
#pragma unroll
  for (int k = 0; k < 4; ++k) pen += fmaxf(-sref * cr[k], 0.f);
  return 0.25f * pen;
}

__device__ __forceinline__ float box_loss(const float px[4], const float py[4],
                                          const float tx[4], const float ty[4]) {
  float nx[8], ny[8];
  candidate_axes(px, py, &nx[0], &ny[0]);
  candidate_axes(tx, ty, &nx[4], &ny[4]);

  float g = 0.f;
#pragma unroll
  for (int a = 0; a < 8; ++a) {
    float ax = nx[a], ay = ny[a];
    float q0 = fmaf(px[0], ax, py[0] * ay);
    float q1 = fmaf(px[1], ax, py[1] * ay);
    float q2 = fmaf(px[2], ax, py[2] * ay);
    float q3 = fmaf(px[3], ax, py[3] * ay);
    float r0 = fmaf(tx[0], ax, ty[0] * ay);
    float r1 = fmaf(tx[1], ax, ty[1] * ay);
    float r2 = fmaf(tx[2], ax, ty[2] * ay);
    float r3 = fmaf(tx[3], ax, ty[3] * ay);
    float min1 = fminf(fminf(q0, q1), fminf(q2, q3));
    float max1 = fmaxf(fmaxf(q0, q1), fmaxf(q2, q3));
    float min2 = fminf(fminf(r0, r1), fminf(r2, r3));
    float max2 = fmaxf(fmaxf(r0, r1), fmaxf(r2, r3));
    float inter = fmaxf(fminf(max1, max2) - fmaxf(min1, min2), 0.f);
    float uni = (max1 - min1) + (max2 - min2) - inter;
    float hull = fmaxf(max1, max2) - fminf(min1, min2);
    // inter/uni - (hull-uni)/hull  ==  inter*(1/uni) + uni*(1/hull) - 1
    g += fmaf(inter, fast_rcp(uni), fmaf(uni, fast_rcp(hull), -1.f));
  }
  float giou = 0.125f * g;
  return 0.5f * (1.f - giou) + convexity(px, py);
}

// ---------------------------------------------------------------------------
// Reductions (deterministic: fixed shuffle tree + fixed LDS tree, no atomics)
// ---------------------------------------------------------------------------
__device__ __forceinline__ float wave_reduce(float v) {
#pragma unroll
  for (int o = 16; o > 0; o >>= 1) v += __shfl_down(v, o, 32);
  return v;
}

__global__ __launch_bounds__(NTHR) void mgiou_kernel(const float* __restrict__ pred,
                                                     const float* __restrict__ tgt,
                                                     float* __restrict__ partial, int n,
                                                     int iters) {
  // per-wave double-buffered staging tile: 32 lanes x 16 floats (pred8 + tgt8)
  __shared__ float smem[WAVES][2][32 * 16];
  __shared__ float wsum[WAVES];

  const int lane = threadIdx.x & 31;
  const int wave = threadIdx.x >> 5;
  const int tid0 = blockIdx.x * blockDim.x + threadIdx.x;
  const int stride = gridDim.x * blockDim.x;

  float* slot0 = &smem[wave][0][lane * 16];
  float* slot1 = &smem[wave][1][lane * 16];
  // loop-invariant LDS byte offsets (low 32 bits of generic shared pointer)
  const unsigned la0 = (unsigned)(size_t)slot0;
  const unsigned la1 = (unsigned)(size_t)slot1;

  float acc = 0.f;
  {
    int idx0 = tid0 < n ? tid0 : (n - 1);
    stage_box(slot0, la0, pred, tgt, idx0);
  }
  for (int k = 0; k < iters; ++k) {
    int i = tid0 + k * stride;
    bool more = (k + 1 < iters);  // uniform
    if (more) {
      int in_ = i + stride;
      int idxn = in_ < n ? in_ : (n - 1);
      if (k & 1)
        stage_box(slot0, la0, pred, tgt, idxn);
      else
        stage_box(slot1, la1, pred, tgt, idxn);
      WAIT_ASYNC_4();  // current buffer's 4 loads are done (in-order completion)
    } else {
      WAIT_ASYNC_0();
    }
    const float* s = (k & 1) ? slot1 : slot0;
    float px[4], py[4], tx[4], ty[4];
#pragma unroll
    for (int q = 0; q < 4; ++q) {
      px[q] = s[2 * q + 0];
      py[q] = s[2 * q + 1];
      tx[q] = s[8 + 2 * q + 0];
      ty[q] = s[9 + 2 * q];
    }
    float l = box_loss(px, py, tx, ty);
    acc += (i < n) ? l : 0.f;
  }

  float w = wave_reduce(acc);
  if (lane == 0) wsum[wave] = w;
  __syncthreads();
  if (wave == 0) {
    float t = (lane < WAVES) ? wsum[lane] : 0.f;
    t = wave_reduce(t);
    if (lane == 0) partial[blockIdx.x] = t;
  }
}

__global__ __launch_bounds__(NTHR) void reduce_kernel(const float* __restrict__ partial,
                                                      float* __restrict__ out, int nparts,
                                                      float inv_n) {
  __shared__ float wsum[WAVES];
  const int lane = threadIdx.x & 31;
  const int wave = threadIdx.x >> 5;
  float a = 0.f;
  for (int i = threadIdx.x; i < nparts; i += NTHR) a += partial[i];
  float w = wave_reduce(a);
  if (lane == 0) wsum[wave] = w;
  __syncthreads();
  if (wave == 0) {
    float t = (lane < WAVES) ? wsum[lane] : 0.f;
    t = wave_reduce(t);
    if (lane == 0) out[0] = t * inv_n;
  }
}

extern "C" void kernel_launch(void* const* d_in, const int* in_sizes, int n_in,
                              void* d_out, int out_size, void* d_ws, size_t ws_size,
                              hipStream_t stream) {
  const float* pred = (const float*)d_in[0];
  const float* tgt = (const float*)d_in[1];
  float* out = (float*)d_out;
  float* partial = (float*)d_ws;  // NBLK floats of scratch
  int nboxes = in_sizes[0] / 8;   // [B,4,2] f32
  int stride = NBLK * NTHR;
  int iters = (nboxes + stride - 1) / stride;  // host-side: avoids in-kernel int-div
  mgiou_kernel<<<NBLK, NTHR, 0, stream>>>(pred, tgt, partial, nboxes, iters);
  reduce_kernel<<<1, NTHR, 0, stream>>>(partial, out, NBLK, 1.0f / (float)nboxes);
}